// SSVEPFormerTH_8881992368920
// MI455X (gfx1250) — compile-verified
//
#include <hip/hip_runtime.h>
#include <hip/hip_bf16.h>
#include <math.h>

// ---------------- problem constants (from reference) ----------------
#define BATCH   8192
#define CHANS   8
#define TLEN    1024
#define NFFT    1024
#define F0      32
#define SREAL   225          // F1-F0
#define S       450          // real||imag
#define SP      464          // padded S (multiple of 16), pad cols kept at 0
#define FILT    16
#define KW      31
#define KCONV   (FILT*KW)    // 496
#define NTILES  (SP/16)      // 29
#define H1      72
#define NCLS    12
#define FLAT    (FILT*S)     // 7200
#define KH      452          // per-filter K (450 + 2 zero pad), mult of 4
#define NH      80           // padded head cols (72 + 8 zero)
#define EPSLN   1e-5f

typedef __attribute__((ext_vector_type(2))) float v2f;
typedef __attribute__((ext_vector_type(8))) float v8f;

// D = A(16x4 f32) * B(4x16 f32) + C(16x16 f32)  -- CDNA5 WMMA, wave32
__device__ __forceinline__ v8f wmma4(v2f a, v2f b, v8f c) {
  return __builtin_amdgcn_wmma_f32_16x16x4_f32(
      /*neg_a=*/false, a, /*neg_b=*/false, b,
      /*c_mod=*/(short)0, c, /*reuse_a=*/false, /*reuse_b=*/false);
}

// -------- CDNA5 async global->LDS copy (ASYNCcnt path), with fallback ------
#define GLB_AS __attribute__((address_space(1)))
#define LDS_AS __attribute__((address_space(3)))
#if __has_builtin(__builtin_amdgcn_global_load_async_to_lds_b128)
#define HAVE_ASYNC 1
#endif

typedef int vint4 __attribute__((vector_size(16)));   // builtin's pointee type
typedef GLB_AS vint4* gvec4p;
typedef LDS_AS vint4* lvec4p;

__device__ __forceinline__ void async_cp16(const void* gsrc, void* ldst) {
#ifdef HAVE_ASYNC
  __builtin_amdgcn_global_load_async_to_lds_b128(
      (gvec4p)(GLB_AS void*)(void*)gsrc, (lvec4p)(LDS_AS void*)ldst, 0, 0);
#else
  *(float4*)ldst = *(const float4*)gsrc;
#endif
}

__device__ __forceinline__ void async_wait() {
#ifdef HAVE_ASYNC
#if __has_builtin(__builtin_amdgcn_s_wait_asynccnt)
  __builtin_amdgcn_s_wait_asynccnt(0);
#else
  asm volatile("s_wait_asynccnt 0" ::: "memory");
#endif
#endif
}

__device__ __forceinline__ float wave_sum(float v) {
#pragma unroll
  for (int off = 16; off > 0; off >>= 1) v += __shfl_xor(v, off, 32);
  return v;
}

__device__ __forceinline__ float gelu_exact(float x) {
  return 0.5f * x * (1.0f + erff(x * 0.70710678118654752440f));
}

// ---------------- k0: DFT basis table [NFFT x SP] ----------------
// basis[t, s] = cos(2*pi*(F0+s)*t/NFFT)/NFFT            s <  225
//             = -sin(2*pi*(F0+s-225)*t/NFFT)/NFFT       225 <= s < 450
__global__ __launch_bounds__(256) void k_basis(float* __restrict__ basis) {
  int idx = blockIdx.x * blockDim.x + threadIdx.x;
  if (idx >= NFFT * SP) return;
  int t = idx / SP, s = idx - t * SP;
  const float W = 6.2831853071795864769f / (float)NFFT;
  float v = 0.0f;
  if (s < SREAL) {
    int ph = ((F0 + s) * t) & (NFFT - 1);          // exact integer phase reduction
    v = cosf((float)ph * W) * (1.0f / (float)NFFT);
  } else if (s < S) {
    int ph = ((F0 + s - SREAL) * t) & (NFFT - 1);
    v = -sinf((float)ph * W) * (1.0f / (float)NFFT);
  }
  basis[idx] = v;
}

// ------- k0b: transpose + pad head weights: w1t[fi*452+s][80] --------------
__global__ __launch_bounds__(256) void k_w1t(const float* __restrict__ w1,
                                             float* __restrict__ w1t) {
  int idx = blockIdx.x * blockDim.x + threadIdx.x;
  if (idx >= FILT * KH * NH) return;
  int k = idx / NH, col = idx - k * NH;
  int fi = k / KH, s = k - fi * KH;
  w1t[idx] = (col < H1 && s < S) ? w1[(size_t)col * FLAT + fi * S + s] : 0.0f;
}

// ---------------- k1: DFT GEMM  dft[65536, SP] = x[65536,1024] @ basis -----
// 8 waves/block share one M-tile; A K-chunks staged via async-to-LDS.
#define ASTRIDE 132                                 // 132 % 64 = 4 -> no bank conflicts
__global__ __launch_bounds__(256) void k_dft(const float* __restrict__ X,
                                             const float* __restrict__ basis,
                                             float* __restrict__ dft) {
  __shared__ float As[16 * ASTRIDE];                // 16 x 128 K-chunk (padded)
  int tm = blockIdx.x >> 2, grp = blockIdx.x & 3;
  int tid = threadIdx.x, wv = tid >> 5, lane = tid & 31;
  int half = lane >> 4, l16 = lane & 15;
  int tn = grp * 8 + wv;
  bool act = (tn < NTILES);
  int col = act ? tn * 16 + l16 : l16;              // clamp inactive: safe reads
  const float4* Xb = (const float4*)(X + (size_t)tm * 16 * TLEN);
  v8f acc = {};
  for (int kc = 0; kc < TLEN / 128; ++kc) {
    for (int i = tid; i < 16 * 32; i += 256) {      // 512 float4, coalesced
      int r = i >> 5, j = i & 31;
      async_cp16(Xb + (size_t)r * (TLEN / 4) + kc * 32 + j,
                 &As[r * ASTRIDE + (j << 2)]);
    }
    async_wait();
    __syncthreads();
    int kbase = kc * 128;
#pragma unroll 4
    for (int kk = 0; kk < 32; ++kk) {
      int lk = kk * 4 + (half << 1);
      v2f a = *(const v2f*)&As[l16 * ASTRIDE + lk]; // A: K={0,1}+2*half
      int kr = kbase + lk;                          // B: V0 rows {K0,K2}, V1 {K1,K3}
      v2f b; b.x = basis[(size_t)kr * SP + col];
             b.y = basis[(size_t)(kr + 1) * SP + col];
      acc = wmma4(a, b, acc);
    }
    __syncthreads();
  }
  if (act) {
#pragma unroll
    for (int v = 0; v < 8; ++v) {                   // D: VGPR v -> M=v (+8 upper half)
      int m = v + (half << 3);
      dft[(size_t)(tm * 16 + m) * SP + col] = acc[v];
    }
  }
}

// ---------------- k2: ChComb(8->16) + LN + GELU, one WG per batch ----------
__global__ __launch_bounds__(512) void k_chcomb(const float* __restrict__ dft,
                                                const float* __restrict__ ccw,
                                                const float* __restrict__ ccb,
                                                const float* __restrict__ lnw,
                                                const float* __restrict__ lnb,
                                                float* __restrict__ hout) {
  __shared__ float ch[CHANS * SP];                  // 14.8 KB
  int b = blockIdx.x, tid = threadIdx.x;
  const float4* src4 = (const float4*)(dft + (size_t)b * CHANS * SP);
  for (int i = tid; i < CHANS * SP / 4; i += 512)
    async_cp16(src4 + i, (float4*)ch + i);
  async_wait();
  __syncthreads();
  int o = tid >> 5, lane = tid & 31;                // wave o = output filter
  float w[CHANS];
#pragma unroll
  for (int c = 0; c < CHANS; ++c) w[c] = ccw[o * CHANS + c];
  float bias = ccb[o];
  float vals[15];
  float sum = 0.f, sq = 0.f; int n = 0;
  for (int s = lane; s < S; s += 32) {
    float a = bias;
#pragma unroll
    for (int c = 0; c < CHANS; ++c) a += ch[c * SP + s] * w[c];
    vals[n++] = a; sum += a; sq += a * a;
  }
  sum = wave_sum(sum); sq = wave_sum(sq);
  float mean = sum * (1.0f / S);
  float rstd = rsqrtf(sq * (1.0f / S) - mean * mean + EPSLN);
  float* out = hout + ((size_t)b * FILT + o) * SP;
  n = 0;
  for (int s = lane; s < S; s += 32)
    out[s] = gelu_exact((vals[n++] - mean) * rstd * lnw[s] + lnb[s]);
  for (int s = S + lane; s < SP; s += 32) out[s] = 0.0f;   // keep pads zero
}

// ---------------- k3: fused encoder block, one WG (16 waves) per batch -----
#define CWSTRIDE 508                     // 508 % 64 = 60 -> conflict-free A fetch
#define SM_HS  0                         // hs [16][464]
#define SM_T1  (16 * SP)                 // t1 [16][512], halo offset 15
#define SM_CW  (SM_T1 + 16 * 512)        // cwl[16][508]
#define SM_YB  (SM_CW + 16 * CWSTRIDE)   // yb [16][464]
#define SM_FLOATS (SM_YB + 16 * SP)      // 31168 floats = 124,672 B dynamic LDS

__global__ __launch_bounds__(512) void k_encoder(
    const float* __restrict__ hin,
    const float* __restrict__ ln1w, const float* __restrict__ ln1b,
    const float* __restrict__ cw,   const float* __restrict__ cb,
    const float* __restrict__ ln2w, const float* __restrict__ ln2b,
    const float* __restrict__ ln3w, const float* __restrict__ ln3b,
    const float* __restrict__ pw,   const float* __restrict__ pb,
    float* __restrict__ hout) {
  extern __shared__ float sm[];
  float* hs  = sm + SM_HS;
  float* t1  = sm + SM_T1;
  float* cwl = sm + SM_CW;
  float* yb  = sm + SM_YB;
  int b = blockIdx.x, tid = threadIdx.x, wv = tid >> 5, lane = tid & 31;
  const size_t base = (size_t)b * FILT * SP;
  const float4* hin4 = (const float4*)(hin + base);
  for (int i = tid; i < FILT * SP / 4; i += 512)            // state slab
    async_cp16(hin4 + i, (float4*)hs + i);
  for (int i = tid; i < FILT * KCONV / 4; i += 512) {       // conv weights
    int o = i / (KCONV / 4), j = i - o * (KCONV / 4);
    async_cp16((const float4*)cw + i, &cwl[o * CWSTRIDE + (j << 2)]);
  }
  for (int i = tid; i < FILT * 512; i += 512) t1[i] = 0.0f; // halos/pads
  async_wait();
  __syncthreads();

  // ---- LN1 into t1 (halo offset 15) ----
  {
    float vals[15]; float sum = 0.f, sq = 0.f; int n = 0;
    for (int s = lane; s < S; s += 32) {
      float v = hs[wv * SP + s]; vals[n++] = v; sum += v; sq += v * v;
    }
    sum = wave_sum(sum); sq = wave_sum(sq);
    float mean = sum * (1.0f / S);
    float rstd = rsqrtf(sq * (1.0f / S) - mean * mean + EPSLN);
    n = 0;
    for (int s = lane; s < S; s += 32)
      t1[wv * 512 + 15 + s] = (vals[n++] - mean) * rstd * ln1w[s] + ln1b[s];
  }
  __syncthreads();

  // ---- conv (KW=31, SAME) as WMMA GEMM: [16 x 496] @ [496 x 16] tiles ----
  {
    int half = lane >> 4, l16 = lane & 15;
    for (int tile = wv; tile < NTILES; tile += 16) {
      int scol = tile * 16 + l16;
      v8f acc = {};
#pragma unroll 4
      for (int kk = 0; kk < KCONV / 4; ++kk) {
        int kr = kk * 4 + (half << 1);
        v2f a = *(const v2f*)&cwl[l16 * CWSTRIDE + kr];   // A rows = out filter
        int i0 = kr / KW,       r0 = kr - i0 * KW;
        int i1 = (kr + 1) / KW, r1 = (kr + 1) - i1 * KW;
        v2f bb; bb.x = t1[i0 * 512 + scol + r0];          // halo = free SAME pad
                bb.y = t1[i1 * 512 + scol + r1];
        acc = wmma4(a, bb, acc);
      }
#pragma unroll
      for (int v = 0; v < 8; ++v) {
        int o = v + (half << 3);
        yb[o * SP + scol] = acc[v] + cb[o];
      }
    }
  }
  __syncthreads();

  // ---- LN2 + GELU + residual into hs ----
  {
    float vals[15]; float sum = 0.f, sq = 0.f; int n = 0;
    for (int s = lane; s < S; s += 32) {
      float v = yb[wv * SP + s]; vals[n++] = v; sum += v; sq += v * v;
    }
    sum = wave_sum(sum); sq = wave_sum(sq);
    float mean = sum * (1.0f / S);
    float rstd = rsqrtf(sq * (1.0f / S) - mean * mean + EPSLN);
    n = 0;
    for (int s = lane; s < S; s += 32) {
      float g = gelu_exact((vals[n++] - mean) * rstd * ln2w[s] + ln2b[s]);
      hs[wv * SP + s] += g;
    }
  }
  __syncthreads();

  // ---- LN3 into t1 (offset 0 reuse) ----
  {
    float vals[15]; float sum = 0.f, sq = 0.f; int n = 0;
    for (int s = lane; s < S; s += 32) {
      float v = hs[wv * SP + s]; vals[n++] = v; sum += v; sq += v * v;
    }
    sum = wave_sum(sum); sq = wave_sum(sq);
    float mean = sum * (1.0f / S);
    float rstd = rsqrtf(sq * (1.0f / S) - mean * mean + EPSLN);
    n = 0;
    for (int s = lane; s < S; s += 32)
      t1[wv * 512 + s] = (vals[n++] - mean) * rstd * ln3w[s] + ln3b[s];
  }
  __syncthreads();

  // ---- proj: out[i,s] = sum_c t3[c,i]*pw[s,c] + pb[s] + h'[i,s] ----
  {
    float col[FILT];
#pragma unroll
    for (int c = 0; c < FILT; ++c) col[c] = t1[c * 512 + wv];
    for (int s = lane; s < S; s += 32) {
      float a = pb[s];
#pragma unroll
      for (int c = 0; c < FILT; ++c) a += col[c] * pw[s * FILT + c];
      hout[base + wv * SP + s] = a + hs[wv * SP + s];
    }
    for (int s = S + lane; s < SP; s += 32) hout[base + wv * SP + s] = 0.0f;
  }
}

// ---------------- k4: head GEMM y1[B,72] = h2[B,7200] @ w1^T + b1 ----------
// one block per 16-batch M-tile; A fi-slabs staged async; 5 waves = 5 N-tiles.
#define HSTRIDE 468                      // 468 % 64 = 20 -> conflict-free A fetch
__global__ __launch_bounds__(256) void k_head1(const float* __restrict__ h,
                                               const float* __restrict__ w1t,
                                               const float* __restrict__ b1,
                                               float* __restrict__ y1) {
  __shared__ float As[16 * HSTRIDE];     // 29.9 KB
  int tm = blockIdx.x;
  int tid = threadIdx.x, wv = tid >> 5, lane = tid & 31;
  int half = lane >> 4, l16 = lane & 15;
  int col = wv * 16 + l16;               // < 80 for the 5 active waves
  const float4* Hb = (const float4*)(h + (size_t)tm * 16 * FILT * SP);
  v8f acc = {};
  for (int fi = 0; fi < FILT; ++fi) {
    for (int i = tid; i < 16 * (SP / 4); i += 256) {        // 1856 float4
      int r = i / (SP / 4), j = i - r * (SP / 4);
      async_cp16(Hb + (size_t)r * (FILT * SP / 4) + fi * (SP / 4) + j,
                 &As[r * HSTRIDE + (j << 2)]);
    }
    async_wait();
    __syncthreads();
    if (wv < 5) {
      const float* wk = w1t + (size_t)fi * KH * NH + col;   // [k][80], zero-padded
#pragma unroll 4
      for (int ss = 0; ss < KH / 4; ++ss) {                 // 113 steps
        int lk = ss * 4 + (half << 1);
        v2f a = *(const v2f*)&As[l16 * HSTRIDE + lk];
        v2f b; b.x = wk[(size_t)lk * NH];
               b.y = wk[(size_t)(lk + 1) * NH];
        acc = wmma4(a, b, acc);
      }
    }
    __syncthreads();
  }
  if (wv < 5 && col < H1) {
#pragma unroll
    for (int v = 0; v < 8; ++v) {
      int m = v + (half << 3);
      y1[(size_t)(tm * 16 + m) * H1 + col] = acc[v] + b1[col];
    }
  }
}

// ---------------- k5: LN(72) + GELU + 12-class matvec, one wave per row ----
__global__ __launch_bounds__(256) void k_head2(const float* __restrict__ y1,
                                               const float* __restrict__ lnw,
                                               const float* __restrict__ lnb,
                                               const float* __restrict__ w2,
                                               const float* __restrict__ b2,
                                               float* __restrict__ out) {
  int b = blockIdx.x * 8 + (threadIdx.x >> 5);
  int lane = threadIdx.x & 31;
  const float* row = y1 + (size_t)b * H1;
  float v0 = row[lane];
  float v1 = row[lane + 32];
  float v2 = (lane < H1 - 64) ? row[lane + 64] : 0.0f;
  float sum = wave_sum(v0 + v1 + v2);
  float sq  = wave_sum(v0 * v0 + v1 * v1 + v2 * v2);
  float mean = sum * (1.0f / H1);
  float rstd = rsqrtf(sq * (1.0f / H1) - mean * mean + EPSLN);
  float g0 = gelu_exact((v0 - mean) * rstd * lnw[lane]      + lnb[lane]);
  float g1 = gelu_exact((v1 - mean) * rstd * lnw[lane + 32] + lnb[lane + 32]);
  float g2 = (lane < H1 - 64)
           ? gelu_exact((v2 - mean) * rstd * lnw[lane + 64] + lnb[lane + 64]) : 0.0f;
#pragma unroll
  for (int c = 0; c < NCLS; ++c) {
    const float* w = w2 + c * H1;
    float p = g0 * w[lane] + g1 * w[lane + 32];
    if (lane < H1 - 64) p += g2 * w[lane + 64];
    p = wave_sum(p);
    if (lane == 0) out[(size_t)b * NCLS + c] = p + b2[c];
  }
}

// ---------------- launch ----------------
extern "C" void kernel_launch(void* const* d_in, const int* in_sizes, int n_in,
                              void* d_out, int out_size, void* d_ws, size_t ws_size,
                              hipStream_t stream) {
  (void)in_sizes; (void)n_in; (void)out_size; (void)ws_size;
  const float* x      = (const float*)d_in[0];
  const float* cc_w   = (const float*)d_in[1];
  const float* cc_b   = (const float*)d_in[2];
  const float* cc_lnw = (const float*)d_in[3];
  const float* cc_lnb = (const float*)d_in[4];
  // e1: ln1w,ln1b,ln2w,ln2b,ln3w,ln3b,convw,convb,projw,projb  (5..14)
  // e2: same                                                    (15..24)
  const float* h_w1  = (const float*)d_in[25];
  const float* h_b1  = (const float*)d_in[26];
  const float* h_lnw = (const float*)d_in[27];
  const float* h_lnb = (const float*)d_in[28];
  const float* h_w2  = (const float*)d_in[29];
  const float* h_b2  = (const float*)d_in[30];

  float* ws = (float*)d_ws;
  const size_t basisOff = 0;
  const size_t hAOff    = basisOff + (size_t)NFFT * SP;        //   1.9 MB
  const size_t hBOff    = hAOff + (size_t)BATCH * FILT * SP;   // 243.3 MB each
  const size_t dftOff   = hBOff;   // dft [65536,SP] aliases hB (dead before enc1 writes)
  const size_t y1Off    = hBOff + (size_t)BATCH * FILT * SP;
  const size_t w1tOff   = y1Off + (size_t)BATCH * H1;          //   2.3 MB
  float* basis = ws + basisOff;
  float* hA    = ws + hAOff;
  float* hB    = ws + hBOff;
  float* dft   = ws + dftOff;
  float* y1    = ws + y1Off;
  float* w1t   = ws + w1tOff;

  k_basis<<<(NFFT * SP + 255) / 256, 256, 0, stream>>>(basis);
  k_w1t<<<(FILT * KH * NH + 255) / 256, 256, 0, stream>>>(h_w1, w1t);
  k_dft<<<(BATCH * CHANS / 16) * 4, 256, 0, stream>>>(x, basis, dft);
  k_chcomb<<<BATCH, 512, 0, stream>>>(dft, cc_w, cc_b, cc_lnw, cc_lnb, hA);

  const size_t smBytes = (size_t)SM_FLOATS * sizeof(float);    // ~125 KB dynamic LDS
  k_encoder<<<BATCH, 512, smBytes, stream>>>(hA,
      (const float*)d_in[5],  (const float*)d_in[6],
      (const float*)d_in[11], (const float*)d_in[12],
      (const float*)d_in[7],  (const float*)d_in[8],
      (const float*)d_in[9],  (const float*)d_in[10],
      (const float*)d_in[13], (const float*)d_in[14], hB);
  k_encoder<<<BATCH, 512, smBytes, stream>>>(hB,
      (const float*)d_in[15], (const float*)d_in[16],
      (const float*)d_in[21], (const float*)d_in[22],
      (const float*)d_in[17], (const float*)d_in[18],
      (const float*)d_in[19], (const float*)d_in[20],
      (const float*)d_in[23], (const float*)d_in[24], hA);

  k_head1<<<BATCH / 16, 256, 0, stream>>>(hA, w1t, h_b1, y1);
  k_head2<<<BATCH / 8, 256, 0, stream>>>(y1, h_lnw, h_lnb, h_w2, h_b2, (float*)d_out);
}